// PillarEncoder_29583734735023
// MI455X (gfx1250) — compile-verified
//
#include <hip/hip_runtime.h>

typedef __attribute__((ext_vector_type(2))) float v2f;
typedef __attribute__((ext_vector_type(8))) float v8f;

#define NPTS   32
#define OUT_C  64
#define XL     432
#define YL     496
#define VXc    0.16f
#define VYc    0.16f
#define XOFF   0.08f
#define YOFF   (-39.60f)
#define BN_EPSf 1e-3f

// ws layout (floats): [0..9) = S, [16..61) = sym M (i<=j), [64..128) = a_c, [128..192) = b_c

// ---------------- Pass 0: zero canvas + stats scratch ----------------
__global__ void zero_out_kernel(float4* __restrict__ o4, float* __restrict__ o,
                                int n4, int ntot) {
    int i = blockIdx.x * blockDim.x + threadIdx.x;
    if (i < n4) o4[i] = make_float4(0.f, 0.f, 0.f, 0.f);
    if (i == 0) for (int t = n4 * 4; t < ntot; ++t) o[t] = 0.f;
}

__global__ void zero_ws_kernel(float* __restrict__ ws) {
    int i = threadIdx.x;
    if (i < 64) ws[i] = 0.f;
}

// ---------------- Pass 1: reduce S (9) and M (45) over all P*NPTS points ----------------
__global__ __launch_bounds__(256) void stats_kernel(
    const float4* __restrict__ pts, const long long* __restrict__ coors,
    const long long* __restrict__ npp, float* __restrict__ ws, int nPillars) {
    int lane   = threadIdx.x & 31;
    int wave   = (blockIdx.x * blockDim.x + threadIdx.x) >> 5;
    int nWaves = (gridDim.x * blockDim.x) >> 5;

    float accS[9], accM[45];
#pragma unroll
    for (int i = 0; i < 9; ++i)  accS[i] = 0.f;
#pragma unroll
    for (int i = 0; i < 45; ++i) accM[i] = 0.f;

    for (int p = wave; p < nPillars; p += nWaves) {
        float4 pt = pts[p * NPTS + lane];
        // centroid over all 32 (padded) rows, divided by true count (matches reference)
        float sx = pt.x, sy = pt.y, sz = pt.z;
#pragma unroll
        for (int m = 16; m >= 1; m >>= 1) {
            sx += __shfl_xor(sx, m, 32);
            sy += __shfl_xor(sy, m, 32);
            sz += __shfl_xor(sz, m, 32);
        }
        int   n    = (int)npp[p];
        float invn = 1.f / (float)n;
        float cxw  = (float)coors[p * 4 + 1];
        float cyw  = (float)coors[p * 4 + 2];
        float f[9];
        f[0] = pt.x - (cxw * VXc + XOFF);
        f[1] = pt.y - (cyw * VYc + YOFF);
        f[2] = pt.z;
        f[3] = pt.w;
        f[4] = pt.x - sx * invn;
        f[5] = pt.y - sy * invn;
        f[6] = pt.z - sz * invn;
        f[7] = f[0];
        f[8] = f[1];
        float msk = (lane < n) ? 1.f : 0.f;
#pragma unroll
        for (int i = 0; i < 9; ++i) f[i] *= msk;
#pragma unroll
        for (int i = 0; i < 9; ++i) accS[i] += f[i];
        int k = 0;
#pragma unroll
        for (int i = 0; i < 9; ++i)
#pragma unroll
            for (int j = i; j < 9; ++j) accM[k++] += f[i] * f[j];
    }

    // wave32 butterfly reductions, then one atomic per value per wave
#pragma unroll
    for (int i = 0; i < 9; ++i) {
        float v = accS[i];
#pragma unroll
        for (int m = 16; m >= 1; m >>= 1) v += __shfl_xor(v, m, 32);
        accS[i] = v;
    }
#pragma unroll
    for (int i = 0; i < 45; ++i) {
        float v = accM[i];
#pragma unroll
        for (int m = 16; m >= 1; m >>= 1) v += __shfl_xor(v, m, 32);
        accM[i] = v;
    }
    if (lane == 0) {
#pragma unroll
        for (int i = 0; i < 9; ++i)  atomicAdd(&ws[i], accS[i]);
#pragma unroll
        for (int i = 0; i < 45; ++i) atomicAdd(&ws[16 + i], accM[i]);
    }
}

// ---------------- Pass 2: fold BN stats into fused affine y = relu(a*x + b) ----------------
__global__ void bn_finalize_kernel(const float* __restrict__ w,
                                   const float* __restrict__ gamma,
                                   const float* __restrict__ beta,
                                   float* __restrict__ ws, float invN) {
    int c = threadIdx.x;
    if (c >= OUT_C) return;
    float wc[9];
#pragma unroll
    for (int i = 0; i < 9; ++i) wc[i] = w[c * 9 + i];
    float mean = 0.f;
#pragma unroll
    for (int i = 0; i < 9; ++i) mean += wc[i] * ws[i];
    mean *= invN;
    float ex2 = 0.f;
    int k = 0;
#pragma unroll
    for (int i = 0; i < 9; ++i)
#pragma unroll
        for (int j = i; j < 9; ++j) {
            float m = ws[16 + k++];
            ex2 += wc[i] * wc[j] * m * ((i == j) ? 1.f : 2.f);
        }
    ex2 *= invN;
    float var = ex2 - mean * mean;
    float a   = gamma[c] * rsqrtf(var + BN_EPSf);
    float b   = beta[c] - mean * a;
    ws[64 + c]  = a;
    ws[128 + c] = b;
}

// ---------------- Pass 3: WMMA GEMM + maxpool + activate + scatter ----------------
__global__ __launch_bounds__(256) void pillar_main_kernel(
    const float4* __restrict__ pts, const long long* __restrict__ coors,
    const long long* __restrict__ npp, const float* __restrict__ w,
    const float* __restrict__ ws, float* __restrict__ out, int nPillars) {
    __shared__ float sm[8][NPTS][12];  // per-wave feat tile, K padded 9->12

    int lane = threadIdx.x & 31;
    int wv   = threadIdx.x >> 5;
    int l15  = lane & 15;
    int half = lane >> 4;  // 0: K sub 0,1 ; 1: K sub 2,3

    // B fragments (4x16 per k-step) and fused BN coefficients — pillar-invariant
    v2f   bf[4][3];
    float aC[4], bC[4];
#pragma unroll
    for (int nt = 0; nt < 4; ++nt) {
        int c  = nt * 16 + l15;
        aC[nt] = ws[64 + c];
        bC[nt] = ws[128 + c];
#pragma unroll
        for (int ks = 0; ks < 3; ++ks) {
            int i0        = ks * 4 + half * 2;
            bf[nt][ks].x  = (i0 < 9)     ? w[c * 9 + i0]     : 0.f;
            bf[nt][ks].y  = (i0 + 1 < 9) ? w[c * 9 + i0 + 1] : 0.f;
        }
    }

    int p = blockIdx.x * 8 + wv;        // uniform per wave -> EXEC stays all-ones
    if (p >= nPillars) return;

    float4 pt = pts[p * NPTS + lane];
    float sx = pt.x, sy = pt.y, sz = pt.z;
#pragma unroll
    for (int m = 16; m >= 1; m >>= 1) {
        sx += __shfl_xor(sx, m, 32);
        sy += __shfl_xor(sy, m, 32);
        sz += __shfl_xor(sz, m, 32);
    }
    int       n    = (int)npp[p];
    float     invn = 1.f / (float)n;
    long long cb   = coors[p * 4 + 0];
    long long cx   = coors[p * 4 + 1];
    long long cy   = coors[p * 4 + 2];

    float f[9];
    f[0] = pt.x - ((float)cx * VXc + XOFF);
    f[1] = pt.y - ((float)cy * VYc + YOFF);
    f[2] = pt.z;
    f[3] = pt.w;
    f[4] = pt.x - sx * invn;
    f[5] = pt.y - sy * invn;
    f[6] = pt.z - sz * invn;
    f[7] = f[0];
    f[8] = f[1];
    float msk = (lane < n) ? 1.f : 0.f;
#pragma unroll
    for (int i = 0; i < 9; ++i) sm[wv][lane][i] = f[i] * msk;
    sm[wv][lane][9]  = 0.f;
    sm[wv][lane][10] = 0.f;
    sm[wv][lane][11] = 0.f;

    // same-wave DS store -> DS load ordering (explicit CDNA5 split counter wait)
    asm volatile("s_wait_dscnt 0" ::: "memory");

    float pmax[4] = {-3.0e38f, -3.0e38f, -3.0e38f, -3.0e38f};
#pragma unroll
    for (int mt = 0; mt < 2; ++mt) {
        // A fragments: 16x4 f32, lane L<16 holds K 0,1 of row L; L>=16 holds K 2,3
        v2f af[3];
        int r = mt * 16 + l15;
#pragma unroll
        for (int ks = 0; ks < 3; ++ks) {
            int c0   = ks * 4 + half * 2;
            af[ks].x = sm[wv][r][c0];
            af[ks].y = sm[wv][r][c0 + 1];
        }
#pragma unroll
        for (int nt = 0; nt < 4; ++nt) {
            v8f acc = {0.f, 0.f, 0.f, 0.f, 0.f, 0.f, 0.f, 0.f};
#pragma unroll
            for (int ks = 0; ks < 3; ++ks)
                acc = __builtin_amdgcn_wmma_f32_16x16x4_f32(
                    false, af[ks], false, bf[nt][ks], (short)0, acc, false, false);
            float mx = acc[0];
#pragma unroll
            for (int e = 1; e < 8; ++e) mx = fmaxf(mx, acc[e]);
            pmax[nt] = fmaxf(pmax[nt], mx);
        }
    }

    // finish point-dim max across lane^16 (rows +8), activate, scatter to canvas
#pragma unroll
    for (int nt = 0; nt < 4; ++nt) {
        float v = fmaxf(pmax[nt], __shfl_xor(pmax[nt], 16, 32));
        v       = fmaxf(aC[nt] * v + bC[nt], 0.f);
        if (lane < 16) {
            int    c   = nt * 16 + l15;
            size_t idx = (((size_t)cb * OUT_C + (size_t)c) * YL + (size_t)cy) * XL + (size_t)cx;
            out[idx]   = v;
        }
    }
}

extern "C" void kernel_launch(void* const* d_in, const int* in_sizes, int n_in,
                              void* d_out, int out_size, void* d_ws, size_t ws_size,
                              hipStream_t stream) {
    const float4*    pts   = (const float4*)d_in[0];
    const long long* coors = (const long long*)d_in[1];
    const long long* npp   = (const long long*)d_in[2];
    const float*     w     = (const float*)d_in[3];
    const float*     gamma = (const float*)d_in[4];
    const float*     beta  = (const float*)d_in[5];
    float*           out   = (float*)d_out;
    float*           ws    = (float*)d_ws;

    int nPillars = in_sizes[0] / (NPTS * 4);
    int n4       = out_size / 4;
    float invN   = 1.f / ((float)nPillars * (float)NPTS);

    zero_out_kernel<<<(n4 + 255) / 256, 256, 0, stream>>>((float4*)d_out, out, n4, out_size);
    zero_ws_kernel<<<1, 64, 0, stream>>>(ws);
    stats_kernel<<<256, 256, 0, stream>>>(pts, coors, npp, ws, nPillars);
    bn_finalize_kernel<<<1, 64, 0, stream>>>(w, gamma, beta, ws, invN);
    pillar_main_kernel<<<(nPillars + 7) / 8, 256, 0, stream>>>(pts, coors, npp, w, ws, out, nPillars);
}